// StepSelectorDecoder_72035191488820
// MI455X (gfx1250) — compile-verified
//
#include <hip/hip_runtime.h>
#include <stddef.h>

// Problem constants (from reference): B=32, L=128, H=512, V=32000
#define B_  32
#define H_  512
#define L_  128
#define V_  32000
#define NT_ (V_ / 16)   // 2000 vocab tiles of 16

typedef __attribute__((ext_vector_type(2))) float v2f;
typedef __attribute__((ext_vector_type(4))) float v4f;
typedef __attribute__((ext_vector_type(8))) float v8f;
typedef __attribute__((ext_vector_type(4))) int   v4i_;

// ---------------------------------------------------------------------------
// CDNA5 fp32 WMMA: D(16x16) = A(16x4) * B(4x16) + C, wave32.
// A layout : lanes 0-15 hold M=lane, {v0,v1}={K0,K1}; lanes 16-31 M=lane-16,
//            {v0,v1}={K2,K3}  -> per-lane v2f at K = k + (lane>=16 ? 2 : 0)
// B layout : lanes 0-15 hold N=lane, {v0,v1}={K0,K1}; lanes 16-31 N=lane-16,
//            {v0,v1}={K2,K3}
// C/D      : VGPR r: lanes 0-15 -> (M=r, N=lane); lanes 16-31 -> (M=8+r, N=lane-16)
// ---------------------------------------------------------------------------
#if __has_builtin(__builtin_amdgcn_wmma_f32_16x16x4_f32)
#define WMMA_F32(accv, av, bv)                                                  \
    (accv) = __builtin_amdgcn_wmma_f32_16x16x4_f32(                             \
        /*neg_a=*/false, (av), /*neg_b=*/false, (bv),                           \
        /*c_mod=*/(short)0, (accv), /*reuse_a=*/false, /*reuse_b=*/false)
#else
#define WMMA_F32(accv, av, bv)                                                  \
    asm volatile("v_wmma_f32_16x16x4_f32 %0, %1, %2, %0"                        \
                 : "+v"(accv) : "v"(av), "v"(bv))
#endif

// ---------------------------------------------------------------------------
// gfx1250 async global->LDS copy (ASYNCcnt-tracked); synchronous fallback if
// the toolchain lacks the builtin. 16 bytes per lane per issue.
// Builtin signature (per hipcc diagnostic): (int4 AS1* src, int4 AS3* dst,
// imm i32 offset, imm i32 cpol).
// ---------------------------------------------------------------------------
#if __has_builtin(__builtin_amdgcn_global_load_async_to_lds_b128)
#define HAVE_ASYNC_LDS 1
#define ASYNC_LDS_COPY16(dst, src)                                              \
    __builtin_amdgcn_global_load_async_to_lds_b128(                             \
        (__attribute__((address_space(1))) v4i_*)(src),                         \
        (__attribute__((address_space(3))) v4i_*)(dst), 0, 0)
#else
#define ASYNC_LDS_COPY16(dst, src)  (*(v4f*)(dst) = *(const v4f*)(src))
#endif

__device__ __forceinline__ void wait_async_stage() {
#if defined(HAVE_ASYNC_LDS)
#if __has_builtin(__builtin_amdgcn_s_wait_asynccnt)
    __builtin_amdgcn_s_wait_asynccnt(0);
#else
    asm volatile("s_wait_asynccnt 0x0");
#endif
#endif
    asm volatile("" ::: "memory");   // keep LDS reads after the wait
}

__device__ __forceinline__ float sigmoid_(float x) { return 1.0f / (1.0f + expf(-x)); }

// ---------------------------------------------------------------------------
// Kernel A: gates = emb[tok] @ W_ih^T + b_ih + h @ W_hh^T + b_hh, fused LSTM
// cell update. One wave per (batch-tile, unit-tile).
// Phase 1: async-gather the 16 embedding rows (32 KB) into LDS.
// Phase 2: h-part GEMM (4 gates) from global — overlaps the async gather.
// Phase 3: s_wait_asynccnt, x-part GEMM with A-fragments from LDS.
// Phase 4: bias + LSTM cell, write h_new/c_new.
// grid = (H/16, B/16), block = 32 (one wave32, EXEC all ones for WMMA).
// ---------------------------------------------------------------------------
__global__ __launch_bounds__(32)
void lstm_step_kernel(int t,
                      const int*   __restrict__ tok,
                      const float* __restrict__ emb,
                      const float* __restrict__ hin,
                      const float* __restrict__ cin,
                      const float* __restrict__ W_ih,
                      const float* __restrict__ W_hh,
                      const float* __restrict__ b_ih,
                      const float* __restrict__ b_hh,
                      float* __restrict__ hout,
                      float* __restrict__ cout)
{
    __shared__ float xlds[16 * H_];      // staged x tile: 16 rows x 512 f32 = 32 KB

    const int lane   = threadIdx.x;
    const int nlocal = lane & 15;
    const int half   = lane >> 4;        // 0: lanes 0-15, 1: lanes 16-31
    const int mbase  = blockIdx.y * 16;  // batch tile
    const int nbase  = blockIdx.x * 16;  // hidden-unit tile
    const int koff   = half * 2;         // K sub-offset per A/B layout

    // ---- Phase 1: issue async gather of emb[tok[mbase..mbase+15]] into LDS.
    // 64 issues x (32 lanes x 16B) = 32 KB. Row index uniform per iteration.
    for (int it = 0; it < 64; ++it) {
        const int r   = it >> 2;                       // row in tile (uniform)
        const int tkr = (t == 0) ? 0 : tok[mbase + r]; // scalar (uniform) load
        const int col = ((it & 3) * 32 + lane) * 4;    // float offset, 16B aligned
        const float* src = emb + (size_t)tkr * H_ + col;
        float*       dst = &xlds[r * H_ + col];
        ASYNC_LDS_COPY16(dst, src);
    }

    // A operand (h) and B operands (weight rows) for this wave
    const int arow = mbase + nlocal;
    const float* hrow = hin + (size_t)arow * H_;
    const float* wih[4];
    const float* whh[4];
#pragma unroll
    for (int g = 0; g < 4; ++g) {
        wih[g] = W_ih + (size_t)(g * H_ + nbase + nlocal) * H_;
        whh[g] = W_hh + (size_t)(g * H_ + nbase + nlocal) * H_;
    }

    v8f zero = {};
    v8f acc[4];
#pragma unroll
    for (int g = 0; g < 4; ++g) acc[g] = zero;

    // ---- Phase 2: h @ W_hh^T while the async gather is in flight.
    // Chunked: batch loads into register arrays, then run the WMMA chain.
    for (int k = 0; k < H_; k += 16) {
        v2f ah[4];
#pragma unroll
        for (int i = 0; i < 4; ++i)
            ah[i] = *(const v2f*)(hrow + k + i * 4 + koff);
#pragma unroll
        for (int g = 0; g < 4; ++g) {
            v2f w[4];
#pragma unroll
            for (int i = 0; i < 4; ++i)
                w[i] = *(const v2f*)(whh[g] + k + i * 4 + koff);
#pragma unroll
            for (int i = 0; i < 4; ++i)
                WMMA_F32(acc[g], ah[i], w[i]);
        }
    }

    // ---- Phase 3: x @ W_ih^T with A-fragments from LDS.
    wait_async_stage();
    for (int k = 0; k < H_; k += 16) {
        v2f ax[4];
#pragma unroll
        for (int i = 0; i < 4; ++i)
            ax[i] = *(const v2f*)&xlds[nlocal * H_ + k + i * 4 + koff];
#pragma unroll
        for (int g = 0; g < 4; ++g) {
            v2f w[4];
#pragma unroll
            for (int i = 0; i < 4; ++i)
                w[i] = *(const v2f*)(wih[g] + k + i * 4 + koff);
#pragma unroll
            for (int i = 0; i < 4; ++i)
                WMMA_F32(acc[g], ax[i], w[i]);
        }
    }

    // ---- Phase 4: bias + LSTM cell (gate order i,f,g,o like PyTorch).
    float bias[4];
#pragma unroll
    for (int g = 0; g < 4; ++g)
        bias[g] = b_ih[g * H_ + nbase + nlocal] + b_hh[g * H_ + nbase + nlocal];

    const int ccol = nbase + nlocal;
#pragma unroll
    for (int r = 0; r < 8; ++r) {
        const int crow = mbase + r + half * 8;
        const float ig = sigmoid_(acc[0][r] + bias[0]);
        const float fg = sigmoid_(acc[1][r] + bias[1]);
        const float gg = tanhf  (acc[2][r] + bias[2]);
        const float og = sigmoid_(acc[3][r] + bias[3]);
        const float c_old = cin[(size_t)crow * H_ + ccol];
        const float c_new = fg * c_old + ig * gg;
        const float h_new = og * tanhf(c_new);
        cout[(size_t)crow * H_ + ccol] = c_new;
        hout[(size_t)crow * H_ + ccol] = h_new;
    }
}

// ---------------------------------------------------------------------------
// Kernel B: logits[b, nbase:nbase+16] = h_new[b] . W_fc^T + b_fc, written to
// d_out[b, t, :]; plus per-tile partial argmax via half-wave shuffles.
// grid = (V/16, B/16), block = 32.
// ---------------------------------------------------------------------------
__global__ __launch_bounds__(32)
void logits_kernel(int t,
                   const float* __restrict__ h,
                   const float* __restrict__ W_fc,
                   const float* __restrict__ b_fc,
                   float* __restrict__ out,       // d_out base, [B, L, V]
                   float* __restrict__ pval,      // [B, NT_]
                   int*   __restrict__ pidx)      // [B, NT_]
{
    const int lane   = threadIdx.x;
    const int nlocal = lane & 15;
    const int half   = lane >> 4;
    const int mbase  = blockIdx.y * 16;
    const int ntile  = blockIdx.x;
    const int nbase  = ntile * 16;
    const int koff   = half * 2;

    const int arow = mbase + nlocal;
    const float* hrow = h    + (size_t)arow * H_;
    const float* wrow = W_fc + (size_t)(nbase + nlocal) * H_;

    v8f acc = {};
    // Chunked: 16 clause-able b64 loads, then 8 chained WMMAs per 32-k chunk.
    for (int k = 0; k < H_; k += 32) {
        __builtin_prefetch(wrow + k + 128, 0, 1);   // global_prefetch_b8 ahead of W_fc stream
        v2f a[8], b[8];
#pragma unroll
        for (int i = 0; i < 8; ++i) {
            a[i] = *(const v2f*)(hrow + k + i * 4 + koff);
            b[i] = *(const v2f*)(wrow + k + i * 4 + koff);
        }
#pragma unroll
        for (int i = 0; i < 8; ++i)
            WMMA_F32(acc, a[i], b[i]);
    }

    const float bias = b_fc[nbase + nlocal];
    const int   vcol = nbase + nlocal;
    const size_t rowstride = (size_t)L_ * (size_t)V_;

#pragma unroll
    for (int r = 0; r < 8; ++r) {
        const int brow = mbase + r + half * 8;
        const float v  = acc[r] + bias;
        out[(size_t)brow * rowstride + (size_t)t * V_ + vcol] = v;

        // Reduce max/argmax across the 16 lanes of this half-wave
        float mv = v;
        int   mi = vcol;
#pragma unroll
        for (int off = 8; off >= 1; off >>= 1) {
            float ov = __shfl_xor(mv, off, 16);
            int   oi = __shfl_xor(mi, off, 16);
            if (ov > mv || (ov == mv && oi < mi)) { mv = ov; mi = oi; }
        }
        if (nlocal == 0) {
            pval[(size_t)brow * NT_ + ntile] = mv;
            pidx[(size_t)brow * NT_ + ntile] = mi;
        }
    }
}

// ---------------------------------------------------------------------------
// Kernel C: reduce NT_ partials per batch row -> greedy token (first-occurrence
// tie break, matching jnp.argmax). grid = B, block = 256.
// ---------------------------------------------------------------------------
__global__ __launch_bounds__(256)
void argmax_reduce_kernel(const float* __restrict__ pval,
                          const int*   __restrict__ pidx,
                          int* __restrict__ tok)
{
    __shared__ float sv[256];
    __shared__ int   si[256];
    const int b   = blockIdx.x;
    const int tid = threadIdx.x;

    float best = -3.402823466e38f;
    int   bi   = 0x7fffffff;
    for (int i = tid; i < NT_; i += 256) {
        const float v  = pval[(size_t)b * NT_ + i];
        const int   ix = pidx[(size_t)b * NT_ + i];
        if (v > best || (v == best && ix < bi)) { best = v; bi = ix; }
    }
    sv[tid] = best; si[tid] = bi;
    __syncthreads();
    for (int s = 128; s > 0; s >>= 1) {
        if (tid < s) {
            if (sv[tid + s] > sv[tid] ||
                (sv[tid + s] == sv[tid] && si[tid + s] < si[tid])) {
                sv[tid] = sv[tid + s];
                si[tid] = si[tid + s];
            }
        }
        __syncthreads();
    }
    if (tid == 0) tok[b] = si[0];
}

// Append final (h, c) to d_out tail: [B*L*V] logits, then h [B*H], then c [B*H]
__global__ void copy_state_kernel(const float* __restrict__ hf,
                                  const float* __restrict__ cf,
                                  float* __restrict__ out)
{
    const int i = blockIdx.x * blockDim.x + threadIdx.x;
    const size_t base = (size_t)B_ * (size_t)L_ * (size_t)V_;
    if (i < B_ * H_) {
        out[base + i]           = hf[i];
        out[base + B_ * H_ + i] = cf[i];
    }
}

extern "C" void kernel_launch(void* const* d_in, const int* in_sizes, int n_in,
                              void* d_out, int out_size, void* d_ws, size_t ws_size,
                              hipStream_t stream)
{
    (void)in_sizes; (void)n_in; (void)out_size; (void)ws_size;
    // setup_inputs order: encoder_out(unused), h0, c0, emb, W_ih, W_hh, b_ih, b_hh, W_fc, b_fc
    const float* h0   = (const float*)d_in[1];
    const float* c0   = (const float*)d_in[2];
    const float* emb  = (const float*)d_in[3];
    const float* W_ih = (const float*)d_in[4];
    const float* W_hh = (const float*)d_in[5];
    const float* b_ih = (const float*)d_in[6];
    const float* b_hh = (const float*)d_in[7];
    const float* W_fc = (const float*)d_in[8];
    const float* b_fc = (const float*)d_in[9];
    float* out = (float*)d_out;

    // Workspace carve-up (~0.7 MB)
    char* ws = (char*)d_ws;
    float* hbuf0 = (float*)ws; ws += (size_t)B_ * H_ * sizeof(float);
    float* hbuf1 = (float*)ws; ws += (size_t)B_ * H_ * sizeof(float);
    float* cbuf  = (float*)ws; ws += (size_t)B_ * H_ * sizeof(float);
    int*   tok   = (int*)ws;   ws += (size_t)B_ * sizeof(int);
    float* pval  = (float*)ws; ws += (size_t)B_ * NT_ * sizeof(float);
    int*   pidx  = (int*)ws;   ws += (size_t)B_ * NT_ * sizeof(int);

    float* hb[2] = { hbuf0, hbuf1 };

    const dim3 gridA(H_ / 16, B_ / 16);   // 32 x 2 waves
    const dim3 gridB(NT_,     B_ / 16);   // 2000 x 2 waves

    for (int t = 0; t < L_; ++t) {
        const float* hin  = (t == 0) ? h0 : hb[t & 1];
        const float* cin  = (t == 0) ? c0 : cbuf;
        float*       hnew = hb[(t + 1) & 1];

        lstm_step_kernel<<<gridA, 32, 0, stream>>>(
            t, tok, emb, hin, cin, W_ih, W_hh, b_ih, b_hh, hnew, cbuf);

        logits_kernel<<<gridB, 32, 0, stream>>>(
            t, hnew, W_fc, b_fc, out, pval, pidx);

        if (t + 1 < L_)
            argmax_reduce_kernel<<<B_, 256, 0, stream>>>(pval, pidx, tok);
    }

    copy_state_kernel<<<(B_ * H_ + 255) / 256, 256, 0, stream>>>(
        hb[L_ & 1], cbuf, out);
}